// se_coder_41059887350103
// MI455X (gfx1250) — compile-verified
//
#include <hip/hip_runtime.h>
#include <hip/hip_bf16.h>

// Problem constants
#define NN 4096
#define WW 16
#define LL 8
#define DD 256
#define HH 4

typedef __attribute__((ext_vector_type(2))) float v2f;
typedef __attribute__((ext_vector_type(8))) float v8f;

// ---------------------------------------------------------------------------
// Kernel 1: alpha[n,w,l,h] = softmax_l( leaky_relu( x[n,w,l,:] . attn[:,h] ) )
// One wave32 per (n,w) pair. Lane = l*4 + h after reduction.
// ---------------------------------------------------------------------------
__global__ __launch_bounds__(256) void k1_alpha(const float* __restrict__ x,
                                                const float* __restrict__ attn,
                                                float* __restrict__ alpha) {
  const int wave = threadIdx.x >> 5;
  const int lane = threadIdx.x & 31;
  const int pair = blockIdx.x * 8 + wave;          // n*WW + w, 0..65535
  const float* xp = x + (size_t)pair * (LL * DD);

  // Preload attn rows this lane needs: d = lane*4+j and 128+lane*4+j, j=0..3.
  // attn is [256][4] row-major -> each row is a contiguous float4.
  float am[8][4];
#pragma unroll
  for (int j = 0; j < 4; ++j) {
    float4 a0 = *(const float4*)(attn + (size_t)(lane * 4 + j) * 4);
    am[j][0] = a0.x; am[j][1] = a0.y; am[j][2] = a0.z; am[j][3] = a0.w;
    float4 a1 = *(const float4*)(attn + (size_t)(128 + lane * 4 + j) * 4);
    am[4 + j][0] = a1.x; am[4 + j][1] = a1.y; am[4 + j][2] = a1.z; am[4 + j][3] = a1.w;
  }

  const int l_a = lane >> 2;
  const int h_a = lane & 3;
  float my_ep = 0.0f;

  for (int l = 0; l < LL; ++l) {
    float4 x0 = *(const float4*)(xp + l * DD + lane * 4);
    float4 x1 = *(const float4*)(xp + l * DD + 128 + lane * 4);
    float p[4];
#pragma unroll
    for (int h = 0; h < 4; ++h) {
      p[h] = x0.x * am[0][h] + x0.y * am[1][h] + x0.z * am[2][h] + x0.w * am[3][h]
           + x1.x * am[4][h] + x1.y * am[5][h] + x1.z * am[6][h] + x1.w * am[7][h];
    }
    // full-wave butterfly reduction (wave32)
#pragma unroll
    for (int off = 16; off >= 1; off >>= 1) {
#pragma unroll
      for (int h = 0; h < 4; ++h) p[h] += __shfl_xor(p[h], off, 32);
    }
    if (l == l_a) my_ep = p[h_a];
  }

  // leaky_relu(0.01)
  my_ep = my_ep >= 0.0f ? my_ep : 0.01f * my_ep;

  // softmax across l (lanes differing in bits [4:2]; h fixed in bits [1:0])
  float m = my_ep;
  m = fmaxf(m, __shfl_xor(m, 4, 32));
  m = fmaxf(m, __shfl_xor(m, 8, 32));
  m = fmaxf(m, __shfl_xor(m, 16, 32));
  float e = __expf(my_ep - m);
  float s = e;
  s += __shfl_xor(s, 4, 32);
  s += __shfl_xor(s, 8, 32);
  s += __shfl_xor(s, 16, 32);

  // alpha layout [N,W,L,H]: index = pair*32 + l*4 + h = pair*32 + lane
  alpha[(size_t)pair * 32 + lane] = e / s;
}

// ---------------------------------------------------------------------------
// Kernel 2: fused = relu(alpha_adj @ x), mean over w, store as cat[n, h*256+d].
// Scramble: alpha_adj[h,n,w,l] = alpha[((n&31)<<7)|(w<<3)|l, n>>8, (n>>5)&7, h]
// 64 threads per node (float4 over d), 4 nodes per 256-thread block.
// ---------------------------------------------------------------------------
__global__ __launch_bounds__(256) void k2_fuse(const float* __restrict__ x,
                                               const float* __restrict__ alpha,
                                               float* __restrict__ cat) {
  const int g = threadIdx.x >> 6;                  // node group 0..3
  const int t = threadIdx.x & 63;                  // float4 chunk over d
  const int n = blockIdx.x * 4 + g;

  __shared__ float sC[4][32];                      // [group][h*8 + l]

  const int w_src = n >> 8;
  const int l_src = (n >> 5) & 7;
  const int nbase = (n & 31) << 7;

  float acc[4][4] = {};

  for (int w = 0; w < WW; ++w) {
    __syncthreads();                               // protect prior-iter reads
    if (t < 32) {
      const int l = t >> 2, h = t & 3;
      const int n_src = nbase | (w << 3) | l;
      sC[g][h * 8 + l] =
          alpha[(size_t)((n_src * WW + w_src) * LL + l_src) * HH + h];
    }
    __syncthreads();

    const float* xp = x + (size_t)(n * WW + w) * (LL * DD) + t * 4;
    float xr[8][4];
#pragma unroll
    for (int l = 0; l < LL; ++l) {
      float4 v = *(const float4*)(xp + l * DD);
      xr[l][0] = v.x; xr[l][1] = v.y; xr[l][2] = v.z; xr[l][3] = v.w;
    }
#pragma unroll
    for (int h = 0; h < HH; ++h) {
      float s0 = 0.f, s1 = 0.f, s2 = 0.f, s3 = 0.f;
#pragma unroll
      for (int l = 0; l < LL; ++l) {
        const float c = sC[g][h * 8 + l];
        s0 += c * xr[l][0];
        s1 += c * xr[l][1];
        s2 += c * xr[l][2];
        s3 += c * xr[l][3];
      }
      acc[h][0] += fmaxf(s0, 0.f);
      acc[h][1] += fmaxf(s1, 0.f);
      acc[h][2] += fmaxf(s2, 0.f);
      acc[h][3] += fmaxf(s3, 0.f);
    }
  }

  const float r = 1.0f / (float)WW;
#pragma unroll
  for (int h = 0; h < HH; ++h) {
    float4 o;
    o.x = acc[h][0] * r;
    o.y = acc[h][1] * r;
    o.z = acc[h][2] * r;
    o.w = acc[h][3] * r;
    *(float4*)(cat + (size_t)n * (HH * DD) + h * DD + t * 4) = o;
  }
}

// ---------------------------------------------------------------------------
// Kernel 3: out[4096,256] = cat[4096,1024] @ lin_w^T[1024,256] + lin_b
// WMMA f32 16x16x4, one wave per 16x16 output tile, 256 k-steps.
// A 16x4 frag: lane m=lane&15, {K=kh, K=kh+1} with kh=2*(lane>>4) -> float2 load
// B 4x16 frag: lane n=lane&15, rows {kh, kh+1}                  -> float2 load
// D 16x16: VGPR r -> M = r + 8*(lane>>4), N = lane&15
// ---------------------------------------------------------------------------
__global__ __launch_bounds__(256) void k3_gemm(const float* __restrict__ cat,
                                               const float* __restrict__ lw,
                                               const float* __restrict__ lb,
                                               float* __restrict__ out) {
  const int wave = threadIdx.x >> 5;
  const int lane = threadIdx.x & 31;
  const int tile = blockIdx.x * 8 + wave;          // 4096 tiles = 256 x 16
  const int i0 = (tile >> 4) * 16;                 // row tile
  const int j0 = (tile & 15) * 16;                 // col tile
  const int lm = lane & 15;
  const int kh = (lane >> 4) * 2;                  // 0 or 2

  const float* arow = cat + (size_t)(i0 + lm) * (HH * DD) + kh;
  const float* brow = lw + (size_t)(j0 + lm) * (HH * DD) + kh;
  const float bias = lb[j0 + lm];

  v8f c;
#pragma unroll
  for (int r = 0; r < 8; ++r) c[r] = bias;

  for (int k = 0; k < HH * DD; k += 4) {
    v2f a = *(const v2f*)(arow + k);
    v2f b = *(const v2f*)(brow + k);
    c = __builtin_amdgcn_wmma_f32_16x16x4_f32(
        /*neg_a=*/false, a, /*neg_b=*/false, b,
        /*c_mod=*/(short)0, c, /*reuse_a=*/false, /*reuse_b=*/false);
  }

  const int m0 = i0 + (lane >> 4) * 8;
  float* orow = out + j0 + lm;
#pragma unroll
  for (int r = 0; r < 8; ++r) orow[(size_t)(m0 + r) * DD] = c[r];
}

// ---------------------------------------------------------------------------
extern "C" void kernel_launch(void* const* d_in, const int* in_sizes, int n_in,
                              void* d_out, int out_size, void* d_ws, size_t ws_size,
                              hipStream_t stream) {
  const float* x_walks = (const float*)d_in[0];   // [4096,16,8,256]
  const float* self_attn = (const float*)d_in[1]; // [256,4]
  const float* lin_w = (const float*)d_in[2];     // [256,1024]
  const float* lin_b = (const float*)d_in[3];     // [256]
  float* out = (float*)d_out;                     // [4096,256]

  float* alpha = (float*)d_ws;                    // 2M floats = 8 MB
  float* cat = alpha + (size_t)NN * WW * LL * HH; // 4M floats = 16 MB

  k1_alpha<<<(NN * WW) / 8, 256, 0, stream>>>(x_walks, self_attn, alpha);
  k2_fuse<<<NN / 4, 256, 0, stream>>>(x_walks, alpha, cat);
  k3_gemm<<<(NN / 16) * (DD / 16) / 8, 256, 0, stream>>>(cat, lin_w, lin_b, out);
}